// Transformer_35321811043151
// MI455X (gfx1250) — compile-verified
//
#include <hip/hip_runtime.h>
#include <hip/hip_bf16.h>

typedef __attribute__((ext_vector_type(2))) float v2f;
typedef __attribute__((ext_vector_type(8))) float v8f;

#define NODE_FEAT 128
#define HD 256
#define HEADS 4
#define DH 64
#define NGRAPH 32

// ---------------------------------------------------------------------------
// WMMA f32 GEMM:  Out[M,Nout] = X[M,Kd] @ W[Nout,Kd]^T + b
// One wave computes a 16(M) x 32(N) tile: one A fragment feeds two B
// fragments / two independent accumulators (2x A reuse, 2x WMMA ILP).
// K-steps of 4 via V_WMMA_F32_16X16X4_F32.  M % 16 == 0, Nout % 32 == 0.
// ---------------------------------------------------------------------------
__global__ __launch_bounds__(32)
void wmma_gemm_bias(const float* __restrict__ X, const float* __restrict__ W,
                    const float* __restrict__ b, float* __restrict__ Out,
                    int M, int Kd, int Nout) {
  const int tileM = blockIdx.x;
  const int tileN = blockIdx.y;      // 32-wide N tile
  const int lane  = threadIdx.x;     // 0..31
  const int half  = lane >> 4;       // 0: K={0,1}, 1: K={2,3}
  const int lrow  = lane & 15;

  const float* __restrict__ xrow  = X + (size_t)(tileM * 16 + lrow) * Kd;
  const float* __restrict__ wrow0 = W + (size_t)(tileN * 32 + lrow) * Kd;
  const float* __restrict__ wrow1 = wrow0 + (size_t)16 * Kd;
  const int kb = 2 * half;

  v8f acc0 = {}, acc1 = {};
  for (int kk = 0; kk < Kd; kk += 4) {
    if ((kk & 31) == 0) {
      __builtin_prefetch(xrow  + kk + 64, 0, 1);   // global_prefetch_b8
      __builtin_prefetch(wrow0 + kk + 64, 0, 1);
      __builtin_prefetch(wrow1 + kk + 64, 0, 1);
    }
    v2f a, b0, b1;
    a.x  = xrow[kk + kb];   a.y  = xrow[kk + kb + 1];
    b0.x = wrow0[kk + kb];  b0.y = wrow0[kk + kb + 1];
    b1.x = wrow1[kk + kb];  b1.y = wrow1[kk + kb + 1];
    acc0 = __builtin_amdgcn_wmma_f32_16x16x4_f32(
        false, a, false, b0, (short)0, acc0, false, false);
    acc1 = __builtin_amdgcn_wmma_f32_16x16x4_f32(
        false, a, false, b1, (short)0, acc1, false, false);
  }

  const int col0 = tileN * 32 + lrow;
  const int col1 = col0 + 16;
  const float bv0 = b[col0];
  const float bv1 = b[col1];
#pragma unroll
  for (int r = 0; r < 8; ++r) {
    const int row = tileM * 16 + r + 8 * half;   // C layout: VGPR r -> M=r / r+8
    Out[(size_t)row * Nout + col0] = acc0[r] + bv0;
    Out[(size_t)row * Nout + col1] = acc1[r] + bv1;
  }
}

// ---------------------------------------------------------------------------
// Edge-phase kernels (scatter softmax over incoming edges).
// ---------------------------------------------------------------------------
__device__ __forceinline__ unsigned f2key(float f) {
  unsigned u = __float_as_uint(f);
  return (u & 0x80000000u) ? ~u : (u | 0x80000000u);
}

__global__ void k_edge_logits(const float* __restrict__ Q, const float* __restrict__ Km,
                              const int* __restrict__ src, const int* __restrict__ dst,
                              float* __restrict__ logits, unsigned* __restrict__ mkey, int E) {
  int tid = blockIdx.x * blockDim.x + threadIdx.x;
  if (tid >= E * HEADS) return;
  int e = tid >> 2, h = tid & 3;
  int d = dst[e], s = src[e];
  const float4* qp = reinterpret_cast<const float4*>(Q  + (size_t)d * HD + h * DH);
  const float4* kp = reinterpret_cast<const float4*>(Km + (size_t)s * HD + h * DH);
  float acc = 0.f;
#pragma unroll
  for (int i = 0; i < DH / 4; ++i) {
    float4 a = qp[i], c = kp[i];
    acc += a.x * c.x + a.y * c.y + a.z * c.z + a.w * c.w;
  }
  acc *= 0.125f;   // 1/sqrt(64)
  logits[tid] = acc;
  atomicMax(&mkey[d * HEADS + h], f2key(acc));
}

__global__ void k_fixmax(unsigned* __restrict__ mk, int n) {
  int i = blockIdx.x * blockDim.x + threadIdx.x;
  if (i >= n) return;
  unsigned k = mk[i];
  float f;
  if (k == 0u)                 f = 0.0f;                              // no in-edges
  else if (k & 0x80000000u)    f = __uint_as_float(k & 0x7FFFFFFFu);
  else                         f = __uint_as_float(~k);
  reinterpret_cast<float*>(mk)[i] = f;
}

__global__ void k_expsum(float* __restrict__ logits, const float* __restrict__ mval,
                         float* __restrict__ den, const int* __restrict__ dst, int E) {
  int tid = blockIdx.x * blockDim.x + threadIdx.x;
  if (tid >= E * HEADS) return;
  int e = tid >> 2, h = tid & 3;
  int d = dst[e];
  float ex = __expf(logits[tid] - mval[d * HEADS + h]);
  logits[tid] = ex;                     // in-place: logits -> exp
  atomicAdd(&den[d * HEADS + h], ex);
}

__global__ void k_agg(const float* __restrict__ ex, const float* __restrict__ den,
                      const float* __restrict__ V, const int* __restrict__ src,
                      const int* __restrict__ dst, float* __restrict__ agg, int E) {
  int tid = blockIdx.x * blockDim.x + threadIdx.x;
  if (tid >= E * HEADS) return;
  int e = tid >> 2, h = tid & 3;
  int d = dst[e], s = src[e];
  float alpha = ex[tid] / den[d * HEADS + h];
  const float4* vp = reinterpret_cast<const float4*>(V + (size_t)s * HD + h * DH);
  float* ap = agg + (size_t)d * HD + h * DH;
#pragma unroll
  for (int i = 0; i < DH / 4; ++i) {
    float4 vv = vp[i];
    atomicAdd(ap + 4 * i + 0, alpha * vv.x);
    atomicAdd(ap + 4 * i + 1, alpha * vv.y);
    atomicAdd(ap + 4 * i + 2, alpha * vv.z);
    atomicAdd(ap + 4 * i + 3, alpha * vv.w);
  }
}

__global__ void k_skip_relu(const float* __restrict__ agg, const float* __restrict__ skip,
                            float* __restrict__ out, int n) {
  int i = blockIdx.x * blockDim.x + threadIdx.x;
  if (i < n) out[i] = fmaxf(agg[i] + skip[i], 0.0f);
}

// ---------------------------------------------------------------------------
// Pooling + MLP head (tiny; naive).
// ---------------------------------------------------------------------------
__global__ void k_fill(float* __restrict__ p, float v, size_t n) {
  size_t i = (size_t)blockIdx.x * blockDim.x + threadIdx.x;
  if (i < n) p[i] = v;
}

__global__ void k_cnt(const int* __restrict__ batch, float* __restrict__ cnt, int N) {
  int i = blockIdx.x * blockDim.x + threadIdx.x;
  if (i < N) atomicAdd(&cnt[batch[i]], 1.0f);
}

__global__ void k_pool(const float* __restrict__ h, const int* __restrict__ batch,
                       float* __restrict__ pool, int coloff, int N) {
  int tid = blockIdx.x * blockDim.x + threadIdx.x;
  if (tid >= N * HD) return;
  int n = tid >> 8, c = tid & 255;
  atomicAdd(&pool[batch[n] * (3 * HD) + coloff + c], h[tid]);
}

__global__ void k_pooldiv(float* __restrict__ pool, const float* __restrict__ cnt, int n) {
  int i = blockIdx.x * blockDim.x + threadIdx.x;
  if (i >= n) return;
  int g = i / (3 * HD);
  pool[i] /= fmaxf(cnt[g], 1.0f);
}

__global__ void k_lin(const float* __restrict__ In, const float* __restrict__ W,
                      const float* __restrict__ b, float* __restrict__ Out,
                      int rows, int Kd, int Nout, int do_relu) {
  int tid = blockIdx.x * blockDim.x + threadIdx.x;
  if (tid >= rows * Nout) return;
  int r = tid / Nout, o = tid % Nout;
  const float* in = In + (size_t)r * Kd;
  const float* w  = W + (size_t)o * Kd;
  float acc = b[o];
  for (int k = 0; k < Kd; ++k) acc += in[k] * w[k];
  Out[tid] = do_relu ? fmaxf(acc, 0.0f) : acc;
}

__global__ void k_lin3(const float* __restrict__ In, const float* __restrict__ W,
                       const float* __restrict__ b, float* __restrict__ Out, int rows) {
  int g = blockIdx.x * blockDim.x + threadIdx.x;
  if (g >= rows) return;
  const float* in = In + (size_t)g * 128;
  float acc = b[0];
  for (int k = 0; k < 128; ++k) acc += in[k] * W[k];
  Out[g] = acc;
}

// ---------------------------------------------------------------------------
// Launcher
// ---------------------------------------------------------------------------
extern "C" void kernel_launch(void* const* d_in, const int* in_sizes, int n_in,
                              void* d_out, int out_size, void* d_ws, size_t ws_size,
                              hipStream_t stream) {
  const float* x     = (const float*)d_in[0];
  const int*   ei    = (const int*)d_in[1];
  const int*   batch = (const int*)d_in[2];
  const int N = in_sizes[0] / NODE_FEAT;     // 20000
  const int E = in_sizes[1] / 2;             // 320000
  const int* src = ei;
  const int* dst = ei + E;

  // Workspace carving (256B aligned cursor).
  char* ws = (char*)d_ws;
  size_t cur = 0;
  auto alloc = [&](size_t bytes) -> char* {
    char* p = ws + cur;
    cur = (cur + bytes + 255) & ~(size_t)255;
    return p;
  };
  float* Q    = (float*)alloc((size_t)N * HD * 4);   // reused as AGG
  float* Kb   = (float*)alloc((size_t)N * HD * 4);
  float* V    = (float*)alloc((size_t)N * HD * 4);
  float* SK   = (float*)alloc((size_t)N * HD * 4);
  float* H1   = (float*)alloc((size_t)N * HD * 4);
  float* H2   = (float*)alloc((size_t)N * HD * 4);
  float* H3   = (float*)alloc((size_t)N * HD * 4);
  float* LOG  = (float*)alloc((size_t)E * HEADS * 4);
  float* MX   = (float*)alloc((size_t)N * HEADS * 4);   // keys then floats (in place)
  float* DEN  = (float*)alloc((size_t)N * HEADS * 4);   // contiguous after MX
  float* POOL = (float*)alloc((size_t)NGRAPH * 3 * HD * 4);
  float* CNT  = (float*)alloc((size_t)NGRAPH * 4);      // contiguous after POOL
  float* G1   = (float*)alloc((size_t)NGRAPH * 128 * 4);
  float* G2   = (float*)alloc((size_t)NGRAPH * 128 * 4);
  (void)ws_size; (void)n_in;

  const int T = 256;
  const int EH = E * HEADS;
  float* Hl[3] = {H1, H2, H3};

  const float* Xl = x;
  int Kd = NODE_FEAT;
  for (int L = 0; L < 3; ++L) {
    const int base = 3 + L * 8;
    const float* Wq = (const float*)d_in[base + 0]; const float* bq = (const float*)d_in[base + 1];
    const float* Wk = (const float*)d_in[base + 2]; const float* bk = (const float*)d_in[base + 3];
    const float* Wv = (const float*)d_in[base + 4]; const float* bv = (const float*)d_in[base + 5];
    const float* Ws = (const float*)d_in[base + 6]; const float* bs = (const float*)d_in[base + 7];

    dim3 grid(N / 16, HD / 32);   // 16x32 tile per wave
    wmma_gemm_bias<<<grid, 32, 0, stream>>>(Xl, Wq, bq, Q,  N, Kd, HD);
    wmma_gemm_bias<<<grid, 32, 0, stream>>>(Xl, Wk, bk, Kb, N, Kd, HD);
    wmma_gemm_bias<<<grid, 32, 0, stream>>>(Xl, Wv, bv, V,  N, Kd, HD);
    wmma_gemm_bias<<<grid, 32, 0, stream>>>(Xl, Ws, bs, SK, N, Kd, HD);

    // zero max-keys + denominators (contiguous, 0u < key(-inf))
    k_fill<<<(2 * N * HEADS + T - 1) / T, T, 0, stream>>>(MX, 0.0f, (size_t)2 * N * HEADS);

    k_edge_logits<<<(EH + T - 1) / T, T, 0, stream>>>(Q, Kb, src, dst, LOG, (unsigned*)MX, E);
    k_fixmax<<<(N * HEADS + T - 1) / T, T, 0, stream>>>((unsigned*)MX, N * HEADS);
    k_expsum<<<(EH + T - 1) / T, T, 0, stream>>>(LOG, MX, DEN, dst, E);

    // Q is dead -> reuse as aggregation buffer
    k_fill<<<((size_t)N * HD + T - 1) / T, T, 0, stream>>>(Q, 0.0f, (size_t)N * HD);
    k_agg<<<(EH + T - 1) / T, T, 0, stream>>>(LOG, DEN, V, src, dst, Q, E);
    k_skip_relu<<<((size_t)N * HD + T - 1) / T, T, 0, stream>>>(Q, SK, Hl[L], N * HD);

    Xl = Hl[L];
    Kd = HD;
  }

  // Mean pool (3 layers concatenated) + MLP head.
  k_fill<<<(NGRAPH * 3 * HD + NGRAPH + T - 1) / T, T, 0, stream>>>(
      POOL, 0.0f, (size_t)(NGRAPH * 3 * HD + NGRAPH));
  k_cnt<<<(N + T - 1) / T, T, 0, stream>>>(batch, CNT, N);
  for (int L = 0; L < 3; ++L)
    k_pool<<<((size_t)N * HD + T - 1) / T, T, 0, stream>>>(Hl[L], batch, POOL, L * HD, N);
  k_pooldiv<<<(NGRAPH * 3 * HD + T - 1) / T, T, 0, stream>>>(POOL, CNT, NGRAPH * 3 * HD);

  const float* W1 = (const float*)d_in[27]; const float* b1 = (const float*)d_in[28];
  const float* W2 = (const float*)d_in[29]; const float* b2 = (const float*)d_in[30];
  const float* W3 = (const float*)d_in[31]; const float* b3 = (const float*)d_in[32];
  k_lin<<<(NGRAPH * 128 + T - 1) / T, T, 0, stream>>>(POOL, W1, b1, G1, NGRAPH, 3 * HD, 128, 1);
  k_lin<<<(NGRAPH * 128 + T - 1) / T, T, 0, stream>>>(G1, W2, b2, G2, NGRAPH, 128, 128, 1);
  k_lin3<<<1, NGRAPH, 0, stream>>>(G2, W3, b3, (float*)d_out, NGRAPH);
}